// APPNPBlock_20667382628956
// MI455X (gfx1250) — compile-verified
//
#include <hip/hip_runtime.h>
#include <hip/hip_bf16.h>

// ---------------------------------------------------------------------------
// APPNP (10-hop propagation) + FFN on MI455X (gfx1250, wave32, WMMA).
// Propagation is L2-resident bandwidth/atomic bound; FFN uses fp32 WMMA
// (V_WMMA_F32_16X16X4_F32), 16-row tiles per wave, weights + intermediate
// tile staged in LDS.
// ---------------------------------------------------------------------------

typedef __attribute__((ext_vector_type(2))) float v2f;
typedef __attribute__((ext_vector_type(8))) float v8f;

#define DF    48
#define TSTR  52          // LDS stride for the intermediate tile (bank-conflict pad)
#define ALPHA 0.1f

// ---- degree: deg[dst[e]] += 1 ---------------------------------------------
__global__ void deg_kernel(const int* __restrict__ dst, float* __restrict__ deg, int e) {
    int i = blockIdx.x * blockDim.x + threadIdx.x;
    if (i < e) atomicAdd(&deg[dst[i]], 1.0f);
}

// ---- norm = rsqrt(max(deg,1)) in place ------------------------------------
__global__ void norm_kernel(float* __restrict__ norm, int n) {
    int i = blockIdx.x * blockDim.x + threadIdx.x;
    if (i < n) norm[i] = rsqrtf(fmaxf(norm[i], 1.0f));
}

// ---- S = feat * norm (initial scaled h) -----------------------------------
__global__ void scale0_kernel(const float* __restrict__ feat, const float* __restrict__ norm,
                              float* __restrict__ S, int total) {
    int i = blockIdx.x * blockDim.x + threadIdx.x;
    if (i >= total) return;
    S[i] = feat[i] * norm[i / DF];
}

// ---- edge scatter: G[dst] += S[src], 12 threads/edge, float4 each ---------
__global__ void prop_kernel(const float* __restrict__ S, const int* __restrict__ src,
                            const int* __restrict__ dst, float* __restrict__ G, int nE) {
    int idx = blockIdx.x * blockDim.x + threadIdx.x;
    int e = idx / 12;
    int q = idx - e * 12;
    if (e >= nE) return;
    int s = src[e];
    int d = dst[e];
    const float4 v = ((const float4*)(S + (size_t)s * DF))[q];
    float* gp = G + (size_t)d * DF + q * 4;
    atomicAdd(gp + 0, v.x);
    atomicAdd(gp + 1, v.y);
    atomicAdd(gp + 2, v.z);
    atomicAdd(gp + 3, v.w);
}

// ---- h = (1-a)*G*norm + a*feat ; out = scaleOut ? h*norm : h --------------
__global__ void update_kernel(const float* __restrict__ G, const float* __restrict__ norm,
                              const float* __restrict__ feat, float* __restrict__ out,
                              int total, int scaleOut) {
    int i = blockIdx.x * blockDim.x + threadIdx.x;
    if (i >= total) return;
    float nm = norm[i / DF];
    float h = (1.0f - ALPHA) * (G[i] * nm) + ALPHA * feat[i];
    out[i] = scaleOut ? h * nm : h;
}

// ---- FFN: rst = relu(h@w1+b1)@w2 + b2 + feat, fp32 WMMA -------------------
__global__ __launch_bounds__(256) void ffn_wmma_kernel(
    const float* __restrict__ h,     // [n,48] propagated representation (r)
    const float* __restrict__ feat,  // [n,48]
    const float* __restrict__ w1, const float* __restrict__ b1,
    const float* __restrict__ w2, const float* __restrict__ b2,
    float* __restrict__ out, int n)
{
    __shared__ float w1s[DF * DF];
    __shared__ float w2s[DF * DF];
    __shared__ float b1s[DF];
    __shared__ float b2s[DF];
    __shared__ float tbuf[8 * 16 * TSTR];   // per-wave 16x48 intermediate tile

    for (int i = threadIdx.x; i < DF * DF; i += 256) { w1s[i] = w1[i]; w2s[i] = w2[i]; }
    if (threadIdx.x < DF) { b1s[threadIdx.x] = b1[threadIdx.x]; b2s[threadIdx.x] = b2[threadIdx.x]; }
    __syncthreads();

    const int wave = threadIdx.x >> 5;
    const int lane = threadIdx.x & 31;
    const int tile = blockIdx.x * 8 + wave;
    const int tiles = (n + 15) >> 4;
    if (tile >= tiles) return;            // wave-uniform: EXEC stays all-1s for WMMA

    const int row0 = tile * 16;
    const int lrow = lane & 15;           // A-matrix row held by this lane
    const int koff = (lane >> 4) * 2;     // A/B K sub-offset: lanes 0-15 -> K,K+1; 16-31 -> K+2,K+3
    const int col  = lane & 15;           // B/C/D column held by this lane
    const int mh   = (lane >> 4) * 8;     // C/D row offset for upper half-wave

    int grow = row0 + lrow;
    if (grow >= n) grow = n - 1;          // clamp reads, predicate stores later
    const float* hrow = h + (size_t)grow * DF;

    // ---- GEMM 1: acc = h_tile(16x48) @ w1(48x48), tiled over 3 N-tiles ----
    v8f acc0 = {}, acc1 = {}, acc2 = {};
#pragma unroll
    for (int ks = 0; ks < 12; ++ks) {
        const int k = ks * 4 + koff;
        v2f a; a[0] = hrow[k]; a[1] = hrow[k + 1];
        v2f b;
        b[0] = w1s[k * DF + col];          b[1] = w1s[(k + 1) * DF + col];
        acc0 = __builtin_amdgcn_wmma_f32_16x16x4_f32(false, a, false, b, (short)0, acc0, false, false);
        b[0] = w1s[k * DF + 16 + col];     b[1] = w1s[(k + 1) * DF + 16 + col];
        acc1 = __builtin_amdgcn_wmma_f32_16x16x4_f32(false, a, false, b, (short)0, acc1, false, false);
        b[0] = w1s[k * DF + 32 + col];     b[1] = w1s[(k + 1) * DF + 32 + col];
        acc2 = __builtin_amdgcn_wmma_f32_16x16x4_f32(false, a, false, b, (short)0, acc2, false, false);
    }

    // ---- bias + ReLU, stage tile to LDS (C/D layout -> row major) ---------
    float* tt = &tbuf[wave * 16 * TSTR];
#pragma unroll
    for (int i2 = 0; i2 < 8; ++i2) {
        const int m = i2 + mh;
        tt[m * TSTR + col]      = fmaxf(acc0[i2] + b1s[col],      0.0f);
        tt[m * TSTR + 16 + col] = fmaxf(acc1[i2] + b1s[16 + col], 0.0f);
        tt[m * TSTR + 32 + col] = fmaxf(acc2[i2] + b1s[32 + col], 0.0f);
    }
    // same-wave LDS ops are in-order; no cross-wave sharing of tbuf regions

    // ---- GEMM 2: d = t(16x48) @ w2(48x48) ---------------------------------
    v8f d0 = {}, d1 = {}, d2 = {};
#pragma unroll
    for (int ks = 0; ks < 12; ++ks) {
        const int k = ks * 4 + koff;
        v2f a; a[0] = tt[lrow * TSTR + k]; a[1] = tt[lrow * TSTR + k + 1];
        v2f b;
        b[0] = w2s[k * DF + col];          b[1] = w2s[(k + 1) * DF + col];
        d0 = __builtin_amdgcn_wmma_f32_16x16x4_f32(false, a, false, b, (short)0, d0, false, false);
        b[0] = w2s[k * DF + 16 + col];     b[1] = w2s[(k + 1) * DF + 16 + col];
        d1 = __builtin_amdgcn_wmma_f32_16x16x4_f32(false, a, false, b, (short)0, d1, false, false);
        b[0] = w2s[k * DF + 32 + col];     b[1] = w2s[(k + 1) * DF + 32 + col];
        d2 = __builtin_amdgcn_wmma_f32_16x16x4_f32(false, a, false, b, (short)0, d2, false, false);
    }

    // ---- epilogue: + b2 + residual feat, predicated store -----------------
#pragma unroll
    for (int i2 = 0; i2 < 8; ++i2) {
        const int m = row0 + i2 + mh;
        if (m < n) {
            const float* fr = feat + (size_t)m * DF;
            float* orow = out + (size_t)m * DF;
            orow[col]      = d0[i2] + b2s[col]      + fr[col];
            orow[16 + col] = d1[i2] + b2s[16 + col] + fr[16 + col];
            orow[32 + col] = d2[i2] + b2s[32 + col] + fr[32 + col];
        }
    }
}

// ---------------------------------------------------------------------------
extern "C" void kernel_launch(void* const* d_in, const int* in_sizes, int n_in,
                              void* d_out, int out_size, void* d_ws, size_t ws_size,
                              hipStream_t stream) {
    const float* feat = (const float*)d_in[0];
    const int*   src  = (const int*)  d_in[1];
    const int*   dst  = (const int*)  d_in[2];
    const float* w1   = (const float*)d_in[3];
    const float* b1   = (const float*)d_in[4];
    const float* w2   = (const float*)d_in[5];
    const float* b2   = (const float*)d_in[6];

    const int n = in_sizes[0] / DF;
    const int e = in_sizes[1];
    const int total = n * DF;

    float* rst = (float*)d_out;                  // [n,48] output 0
    float* r   = (float*)d_out + (size_t)total;  // [n,48] output 1 (final h)

    // workspace carve-up: norm[n] | S[n*48] | G[n*48]
    char* ws = (char*)d_ws;
    size_t o = 0;
    float* norm = (float*)(ws + o); o += (((size_t)n * 4) + 255) & ~(size_t)255;
    float* S    = (float*)(ws + o); o += (((size_t)total * 4) + 255) & ~(size_t)255;
    float* G    = (float*)(ws + o);

    const int B = 256;

    // degree -> norm
    hipMemsetAsync(norm, 0, (size_t)n * 4, stream);
    deg_kernel<<<(e + B - 1) / B, B, 0, stream>>>(dst, norm, e);
    norm_kernel<<<(n + B - 1) / B, B, 0, stream>>>(norm, n);

    // S = feat * norm
    scale0_kernel<<<(total + B - 1) / B, B, 0, stream>>>(feat, norm, S, total);

    // 10 propagation hops
    const int propThreads = e * 12;
    for (int hop = 0; hop < 10; ++hop) {
        hipMemsetAsync(G, 0, (size_t)total * 4, stream);
        prop_kernel<<<(propThreads + B - 1) / B, B, 0, stream>>>(S, src, dst, G, e);
        const int last = (hop == 9);
        update_kernel<<<(total + B - 1) / B, B, 0, stream>>>(
            G, norm, feat, last ? r : S, total, last ? 0 : 1);
    }

    // FFN (reads r in-place from d_out second half, writes rst)
    const int tiles = (n + 15) / 16;
    const int blocks = (tiles + 7) / 8;
    ffn_wmma_kernel<<<blocks, 256, 0, stream>>>(r, feat, w1, b1, w2, b2, rst, n);
}